// LowPassFilter_52226802320107
// MI455X (gfx1250) — compile-verified
//
#include <hip/hip_runtime.h>
#include <math.h>

// ---------------------------------------------------------------------------
// Time-varying 9-tap low-pass FIR, MI455X (gfx1250).
//
// out[t] = sum_k filt[t,k] * x[t+4-k] / sum_k filt[t,k]
//   filt[t,k] = 2c * window[k] * sinc(2*c*pi*(k-4)),
//   c = (pi/1000 + alpha*sin(beta*t)) * 8000/2 * pi
//
// Roofline: 16 MB in + 16 MB out = 32 MB @ 23.3 TB/s => ~1.4 us floor.
// Naive math is transcendental-bound (10 sins/output); restructured via
// even-symmetry of window*sinc (4 distinct sincs) + harmonic recurrences
// (one native v_sin/v_cos pair per output) + ONE accurate sincosf per thread
// (sin(beta*t), args up to 4e4 rad, outside v_sin range) amortized over 8
// outputs via angle addition with a uniform sin/cos(beta*r) table.
//
// Data movement: per 2048-output block, stage x[t0-4 .. t0+2059] into LDS via
// GLOBAL_LOAD_ASYNC_TO_LDS_B128 (ASYNCcnt + s_wait_asynccnt); halo reuse is
// LDS-served, each input float crosses the fabric once.
// ---------------------------------------------------------------------------

#define TPB   256
#define OPT   8
#define TILE  (TPB * OPT)      // 2048 outputs per block
#define HALO  4
#define LDSN  (TILE + 16)      // 2064 floats staged (multiple of 4 for b128)

#if defined(__has_builtin)
#  if __has_builtin(__builtin_amdgcn_global_load_async_to_lds_b128)
#    define HAVE_ASYNC_LDS 1
#  endif
#  if __has_builtin(__builtin_amdgcn_s_wait_asynccnt)
#    define HAVE_WAIT_ASYNC 1
#  endif
#endif

typedef int v4i __attribute__((ext_vector_type(4)));
typedef __attribute__((address_space(1))) v4i gv4i_t;   // global int4
typedef __attribute__((address_space(3))) v4i lv4i_t;   // LDS int4

// window[4+j] = sin(pi*(4+j)/8)^2  (f32-pi as in the reference)
#define W1 0.85355339f
#define W2 0.5f
#define W3 0.14644661f
#define W4 7.6427426e-15f   // sin(fl32(pi))^2

__global__ __launch_bounds__(TPB)
void lpf_kernel(const float* __restrict__ x,
                const float* __restrict__ p_alpha,
                const float* __restrict__ p_beta,
                float* __restrict__ out,
                int n, int m)
{
    __shared__ __align__(16) float sx[LDSN];

    const int  tid = threadIdx.x;
    const long t0  = (long)blockIdx.x * TILE;
    const long g0  = t0 - HALO;                  // first staged x index

    const bool interior = (g0 >= 0) && (g0 + LDSN <= (long)n);
    if (interior) {
#if defined(HAVE_ASYNC_LDS)
        const float* gbase = x + g0;             // 16B aligned (t0 % 2048 == 0)
        #pragma unroll 2
        for (int ch = tid; ch < LDSN / 4; ch += TPB) {
            __builtin_amdgcn_global_load_async_to_lds_b128(
                (gv4i_t*)(gbase + 4 * ch), (lv4i_t*)&sx[4 * ch],
                /*offset=*/0, /*cpol=*/0);
        }
#  if defined(HAVE_WAIT_ASYNC)
        __builtin_amdgcn_s_wait_asynccnt(0);
#  else
        asm volatile("s_wait_asynccnt 0" ::: "memory");
#  endif
#else
        const float4* gbase = (const float4*)(x + g0);
        #pragma unroll 2
        for (int ch = tid; ch < LDSN / 4; ch += TPB)
            ((float4*)sx)[ch] = gbase[ch];
#endif
    } else {
        // boundary blocks: guarded fill with zero padding
        for (int i = tid; i < LDSN; i += TPB) {
            long g = g0 + i;
            sx[i] = (g >= 0 && g < (long)n) ? x[g] : 0.0f;
        }
    }
    __syncthreads();

    const long tbase = t0 + (long)tid * OPT;
    if (tbase >= (long)m) return;

    const float alpha = p_alpha[0];
    const float beta  = p_beta[0];
    const float PI_F  = 3.14159274101257324f;

    // accurate sin/cos of beta*t once per thread; angle-add for t+1..t+7
    float s0, c0;
    sincosf(beta * (float)tbase, &s0, &c0);
    // uniform small-angle table sin/cos(beta*r), r = 0..7 (native seed is plenty)
    float SR[OPT], CR[OPT];
    SR[0] = 0.f;
    CR[0] = 1.f;
    const float sb1 = __sinf(beta), cb1 = __cosf(beta);
    #pragma unroll
    for (int r = 1; r < OPT; ++r) {
        SR[r] = SR[r - 1] * cb1 + CR[r - 1] * sb1;
        CR[r] = CR[r - 1] * cb1 - SR[r - 1] * sb1;
    }

    float res[OPT];
    #pragma unroll
    for (int r = 0; r < OPT; ++r) {
        const float sb = s0 * CR[r] + c0 * SR[r];       // sin(beta*(t+r))
        float cc = 0.0031415927f + alpha * sb;          // pi/1000 + a*sin
        cc = cc * 8000.0f;                              // * FS
        cc = cc * 0.5f;                                 // / 2
        cc = cc * PI_F;                                 // * pi

        const float x1 = 2.0f * cc * PI_F;              // sinc arg, j = 1
        const float s1 = __sinf(x1), c1 = __cosf(x1);   // native v_sin/v_cos
        const float s2 = 2.f * s1 * c1;                 // sin(2*x1)
        const float c2 = 1.f - 2.f * s1 * s1;           // cos(2*x1)
        const float s3 = s1 * c2 + c1 * s2;             // sin(3*x1)
        const float s4 = 2.f * s2 * c2;                 // sin(4*x1)

        const float inv  = 1.0f / x1;
        const float twoC = 2.0f * cc;
        const float u4 = twoC;                               // center tap
        const float u5 = twoC * W1 * (s1 * inv);             // |j| = 1
        const float u6 = twoC * W2 * (s2 * (0.5f * inv));    // |j| = 2
        const float u7 = twoC * W3 * (s3 * (0.333333343f * inv)); // |j| = 3
        const float u8 = twoC * W4 * (s4 * (0.25f * inv));   // |j| = 4
        const float S  = u4 + 2.0f * (u5 + u6 + u7 + u8);    // row sum

        const int li = tid * OPT + r + HALO;                 // sx index of x[t]
        const float acc = u4 *  sx[li]
                        + u5 * (sx[li - 1] + sx[li + 1])
                        + u6 * (sx[li - 2] + sx[li + 2])
                        + u7 * (sx[li - 3] + sx[li + 3])
                        + u8 * (sx[li - 4] + sx[li + 4]);
        res[r] = acc / S;
    }

    if (tbase + OPT <= (long)m) {
        // two aligned 16B stores (tbase is a multiple of 8)
        *(float4*)(out + tbase)     = make_float4(res[0], res[1], res[2], res[3]);
        *(float4*)(out + tbase + 4) = make_float4(res[4], res[5], res[6], res[7]);
    } else {
        for (int r = 0; r < OPT; ++r)
            if (tbase + r < (long)m) out[tbase + r] = res[r];
    }
}

extern "C" void kernel_launch(void* const* d_in, const int* in_sizes, int n_in,
                              void* d_out, int out_size, void* d_ws, size_t ws_size,
                              hipStream_t stream) {
    const float* x  = (const float*)d_in[0];
    const float* pa = (const float*)d_in[1];
    const float* pb = (const float*)d_in[2];
    float* out = (float*)d_out;
    const int n = in_sizes[0];
    const int m = out_size;            // n + HS
    const int grid = (m + TILE - 1) / TILE;
    lpf_kernel<<<grid, TPB, 0, stream>>>(x, pa, pb, out, n, m);
}